// GammaE_48945447305870
// MI455X (gfx1250) — compile-verified
//
#include <hip/hip_runtime.h>
#include <math.h>

// ---------------------------------------------------------------------------
// 1-NN via WMMA f32 16x16x4:
//   score[m,n] = pt_sq[m] - 2 * dot(train[n], point[m])
// as A(16x4) x B(4x16):
//   A row m = [px, py, px^2+py^2, 0]
//   B col n = [-2*tx, -2*ty, 1, 0]
// argmin over m per n, then E-lookup; logsumexp handled by a second kernel.
// ---------------------------------------------------------------------------

typedef __attribute__((ext_vector_type(2))) float v2f;
typedef __attribute__((ext_vector_type(8))) float v8f;

#define N_TRAIN   8192
#define GRID_N    200
#define M_PTS     (GRID_N * GRID_N)     // 40000
#define WAVE_SZ   32
#define BLK       256
#define WAVES_PER_BLK (BLK / WAVE_SZ)   // 8
#define N_WAVES   (N_TRAIN / 16)        // 512 waves, 16 train pts each

__global__ __launch_bounds__(BLK)
void nn_wmma_kernel(const float* __restrict__ train,
                    const float* __restrict__ points,
                    const float* __restrict__ E,
                    float* __restrict__ partial)
{
    const int lane = threadIdx.x & (WAVE_SZ - 1);
    const int half = lane >> 4;          // 0: lanes 0-15, 1: lanes 16-31
    const int lm   = lane & 15;
    const int wave = blockIdx.x * WAVES_PER_BLK + (threadIdx.x >> 5);
    const int n0   = wave * 16;          // first train point of this wave

    // B matrix (4x16, K split across lane halves): constant over the M loop.
    const int   n  = n0 + lm;
    const float tx = train[2 * n];
    const float ty = train[2 * n + 1];
    v2f b;
    b[0] = half ? 1.0f : (-2.0f * tx);   // K0 / K2
    b[1] = half ? 0.0f : (-2.0f * ty);   // K1 / K3

    float bestv  = 3.0e38f;
    int   bestid = 0;

    for (int t = 0; t < M_PTS; t += 16) {
        // A matrix tile: 16 grid points, K split across lane halves.
        const int   m    = t + lm;
        const float px   = points[2 * m];
        const float py   = points[2 * m + 1];
        const float ptsq = __builtin_fmaf(px, px, py * py);
        v2f a;
        a[0] = half ? ptsq : px;         // K0 / K2
        a[1] = half ? 0.0f : py;         // K1 / K3

        v8f c = {};
        // D[v]: lanes 0-15 -> (M = t+v,   N = lane)
        //       lanes16-31 -> (M = t+v+8, N = lane-16)
        v8f d = __builtin_amdgcn_wmma_f32_16x16x4_f32(
            /*neg_a=*/false, a, /*neg_b=*/false, b,
            /*c_mod=*/(short)0, c, /*reuse_a=*/false, /*reuse_b=*/false);

#pragma unroll
        for (int v = 0; v < 8; ++v) {
            const float s   = d[v];
            const int   idx = t + v + (half << 3);
            if (s < bestv) { bestv = s; bestid = idx; }  // strict '<' keeps lowest idx
        }
    }

    // Merge the two wave halves (both hold candidates for train point n0+lm).
    const float pv = __shfl_xor(bestv, 16, WAVE_SZ);
    const int   pi = __shfl_xor(bestid, 16, WAVE_SZ);
    if (pv < bestv || (pv == bestv && pi < bestid)) { bestv = pv; bestid = pi; }

    // Lower half contributes E[best]; upper half contributes 0 (avoid double count).
    float e = (half == 0) ? E[bestid] : 0.0f;
#pragma unroll
    for (int s = 16; s >= 1; s >>= 1) e += __shfl_xor(e, s, WAVE_SZ);

    if (lane == 0) partial[wave] = e;    // one deterministic write per wave
}

__global__ __launch_bounds__(1024)
void finalize_kernel(const float* __restrict__ E,
                     const float* __restrict__ partial,
                     float* __restrict__ out)
{
    __shared__ float red[1024];
    const int tid = threadIdx.x;

    // --- max of -E (for stable logsumexp) ---
    float mx = -3.0e38f;
    for (int i = tid; i < M_PTS; i += 1024) mx = fmaxf(mx, -E[i]);
    red[tid] = mx; __syncthreads();
    for (int s = 512; s > 0; s >>= 1) {
        if (tid < s) red[tid] = fmaxf(red[tid], red[tid + s]);
        __syncthreads();
    }
    mx = red[0]; __syncthreads();

    // --- sum exp(-E - mx) ---
    float sm = 0.0f;
    for (int i = tid; i < M_PTS; i += 1024) sm += expf(-E[i] - mx);
    red[tid] = sm; __syncthreads();
    for (int s = 512; s > 0; s >>= 1) {
        if (tid < s) red[tid] += red[tid + s];
        __syncthreads();
    }
    const float lse = mx + logf(red[0]);
    __syncthreads();

    // --- sum of per-wave E[ids] partials ---
    float ps = 0.0f;
    for (int i = tid; i < N_WAVES; i += 1024) ps += partial[i];
    red[tid] = ps; __syncthreads();
    for (int s = 512; s > 0; s >>= 1) {
        if (tid < s) red[tid] += red[tid + s];
        __syncthreads();
    }

    if (tid == 0) {
        const float logZ = lse + logf(0.05f) + logf(0.05f); // + log(DX) + log(DY)
        out[0] = -red[0] / (float)N_TRAIN - logZ;
    }
}

extern "C" void kernel_launch(void* const* d_in, const int* in_sizes, int n_in,
                              void* d_out, int out_size, void* d_ws, size_t ws_size,
                              hipStream_t stream)
{
    const float* train  = (const float*)d_in[0];  // [8192, 2]
    const float* points = (const float*)d_in[1];  // [40000, 2]
    const float* E      = (const float*)d_in[2];  // [40000]
    float*       out    = (float*)d_out;          // scalar
    float*       part   = (float*)d_ws;           // 512 floats of scratch

    nn_wmma_kernel<<<N_WAVES / WAVES_PER_BLK, BLK, 0, stream>>>(train, points, E, part);
    finalize_kernel<<<1, 1024, 0, stream>>>(E, part, out);
}